// RefAtomFeatureEmbedder_80994493268215
// MI455X (gfx1250) — compile-verified
//
#include <hip/hip_runtime.h>
#include <hip/hip_bf16.h>

// ---------------------------------------------------------------------------
// Sizes from the reference
// ---------------------------------------------------------------------------
#define NN      2048          // atoms
#define CREF    389           // feature dim (3+1+1+128+256)
#define KPAD    392           // CREF rounded up to multiple of 4 (WMMA K step)
#define KMAIN   388           // main-loop bound: k<=384 -> max index 387 < 389
#define CATOM   128           // cl channels
#define CPAIR   16            // plm channels
#define LDS_STRIDE 396        // padded row stride for A tile (bank-friendly)

typedef __attribute__((ext_vector_type(2))) float v2f;
typedef __attribute__((ext_vector_type(4))) float v4f;
typedef __attribute__((ext_vector_type(8))) float v8f;

// ---------------------------------------------------------------------------
// Kernel 1: cl = feats @ W_feats + b_feats via V_WMMA_F32_16X16X4_F32
//   grid  = (128 M-tiles, 2),  block = 128 threads (4 wave32)
//   wave w of block (bx, by) computes output tile  M = bx*16..+16,
//                                                  N = (by*4+w)*16..+16
// ---------------------------------------------------------------------------
__global__ __launch_bounds__(128)
void cl_wmma_f32_kernel(const float* __restrict__ pos,
                        const float* __restrict__ charge,
                        const float* __restrict__ mask,
                        const float* __restrict__ elem,
                        const float* __restrict__ names,
                        const float* __restrict__ Wf,   // [389,128] row-major
                        const float* __restrict__ bf,   // [128]
                        float* __restrict__ out)        // [2048,128]
{
    __shared__ float As[16 * LDS_STRIDE];   // 16 rows x 392 (zero-padded) cols

    const int tid   = threadIdx.x;          // 0..127
    const int wave  = tid >> 5;             // 0..3
    const int lane  = tid & 31;             // wave32
    const int mtile = blockIdx.x;           // 0..127
    const int ntile = blockIdx.y * 4 + wave;// 0..7

    // ---- stage gathered A tile (16 x KPAD) into LDS, zero-padded ----
    for (int j = tid; j < 16 * KPAD; j += 128) {
        const int row = j / KPAD;           // 0..15
        const int col = j - row * KPAD;     // 0..391
        const int l   = mtile * 16 + row;   // global atom index
        float v;
        if (col < 3)          v = pos[l * 3 + col];
        else if (col == 3)    v = charge[l];
        else if (col == 4)    v = mask[l];
        else if (col < 133)   v = elem[l * 128 + (col - 5)];
        else if (col < CREF)  v = names[l * 256 + (col - 133)];
        else                  v = 0.0f;     // K padding
        As[row * LDS_STRIDE + col] = v;
    }
    __syncthreads();

    // ---- per-lane WMMA operand indices (ISA 16x16x4 f32 layout) ----
    const int mn   = lane & 15;             // A: M row / B: N col for this lane
    const int koff = (lane >> 4) * 2;       // K sub-offset: lanes16-31 -> K+2
    const int n    = ntile * 16 + mn;       // global output column

    v8f acc = {};
    const float* Arow = &As[mn * LDS_STRIDE];
    const float* Bcol = Wf + n;             // column n, stride CATOM in K

    // ---- main loop: k in [0,384], indices k+koff+1 <= 387 < 389: no guards
    for (int k = 0; k < KMAIN; k += 4) {
        const int k0 = k + koff;
        v2f a;
        a.x = Arow[k0];
        a.y = Arow[k0 + 1];
        v2f b;
        b.x = Bcol[k0 * CATOM];
        b.y = Bcol[(k0 + 1) * CATOM];
        // D = A(16x4) x B(4x16) + C   (fp32 WMMA, wave32)
        acc = __builtin_amdgcn_wmma_f32_16x16x4_f32(
                  false, a, false, b, (short)0, acc, false, false);
    }

    // ---- epilogue: k = 388 covers K 388..391; only K=388 is real ----
    {
        v2f a;
        a.x = Arow[KMAIN + koff];           // LDS zero-padded past 389
        a.y = Arow[KMAIN + koff + 1];
        v2f b;
        b.x = (koff == 0) ? Bcol[KMAIN * CATOM] : 0.0f;  // uniform per half-wave
        b.y = 0.0f;
        acc = __builtin_amdgcn_wmma_f32_16x16x4_f32(
                  false, a, false, b, (short)0, acc, false, false);
    }

    // ---- write D: VGPR r holds M = r + 8*(lane/16), N = lane%16 ----
    const float bias = bf[n];
    const int   mhi  = (lane >> 4) * 8;
    #pragma unroll
    for (int r = 0; r < 8; ++r) {
        const int m = mtile * 16 + mhi + r;
        out[m * CATOM + n] = acc[r] + bias;
    }
}

// ---------------------------------------------------------------------------
// Kernel 2: plm[l][m][c]  (store-bandwidth bound: 268 MB, NT 128-bit stores)
//   grid = (2048/256, 2048), block = 256.  One thread per (l,m) pair.
// ---------------------------------------------------------------------------
__global__ __launch_bounds__(256)
void plm_pair_kernel(const float* __restrict__ pos,
                     const int*   __restrict__ uid,
                     const float* __restrict__ W_off,  // [3,16]
                     const float* __restrict__ b_off,  // [16]
                     const float* __restrict__ W_inv,  // [1,16]
                     const float* __restrict__ b_inv,  // [16]
                     const float* __restrict__ W_vm,   // [1,16]
                     const float* __restrict__ b_vm,   // [16]
                     float* __restrict__ out)          // [2048,2048,16]
{
    __shared__ float w0[CPAIR], w1[CPAIR], w2[CPAIR], wi[CPAIR], wb[CPAIR];
    __shared__ float pl[3];
    __shared__ int   ul;

    const int l = blockIdx.y;
    const int m = blockIdx.x * 256 + threadIdx.x;

    if (threadIdx.x < CPAIR) {
        const int c = threadIdx.x;
        w0[c] = W_off[0 * CPAIR + c];
        w1[c] = W_off[1 * CPAIR + c];
        w2[c] = W_off[2 * CPAIR + c];
        wi[c] = W_inv[c];
        // combined constant term: b_off + b_inv + W_vm + b_vm  (all masked by v)
        wb[c] = b_off[c] + b_inv[c] + W_vm[c] + b_vm[c];
    } else if (threadIdx.x == CPAIR) {
        pl[0] = pos[l * 3 + 0];
        pl[1] = pos[l * 3 + 1];
        pl[2] = pos[l * 3 + 2];
        ul    = uid[l];
    }
    __syncthreads();

    const float d0 = pl[0] - pos[m * 3 + 0];
    const float d1 = pl[1] - pos[m * 3 + 1];
    const float d2 = pl[2] - pos[m * 3 + 2];
    const float v  = (uid[m] == ul) ? 1.0f : 0.0f;
    const float inv = 1.0f / (1.0f + d0 * d0 + d1 * d1 + d2 * d2);

    float r[CPAIR];
    #pragma unroll
    for (int c = 0; c < CPAIR; ++c) {
        r[c] = v * (d0 * w0[c] + d1 * w1[c] + d2 * w2[c] + inv * wi[c] + wb[c]);
    }

    v4f* o = (v4f*)(out + ((size_t)l * NN + m) * CPAIR);
    #pragma unroll
    for (int q = 0; q < 4; ++q) {
        v4f val = { r[4 * q + 0], r[4 * q + 1], r[4 * q + 2], r[4 * q + 3] };
        __builtin_nontemporal_store(val, o + q);   // NT: 268MB stream > L2
    }
}

// ---------------------------------------------------------------------------
// Host-side launch
// ---------------------------------------------------------------------------
extern "C" void kernel_launch(void* const* d_in, const int* in_sizes, int n_in,
                              void* d_out, int out_size, void* d_ws, size_t ws_size,
                              hipStream_t stream) {
    const float* pos    = (const float*)d_in[0];   // [1,2048,3]
    const float* charge = (const float*)d_in[1];   // [1,2048]
    const float* maskp  = (const float*)d_in[2];   // [1,2048]
    const float* elem   = (const float*)d_in[3];   // [1,2048,128]
    const float* names  = (const float*)d_in[4];   // [1,2048,4,64] -> flat 256
    const int*   uid    = (const int*)  d_in[5];   // [1,2048]
    const float* Wf     = (const float*)d_in[6];   // [389,128]
    const float* bf     = (const float*)d_in[7];   // [128]
    const float* W_off  = (const float*)d_in[8];   // [3,16]
    const float* b_off  = (const float*)d_in[9];   // [16]
    const float* W_inv  = (const float*)d_in[10];  // [1,16]
    const float* b_inv  = (const float*)d_in[11];  // [16]
    const float* W_vm   = (const float*)d_in[12];  // [1,16]
    const float* b_vm   = (const float*)d_in[13];  // [16]

    float* out_cl  = (float*)d_out;                      // [2048,128]
    float* out_plm = (float*)d_out + (size_t)NN * CATOM; // [2048,2048,16]

    // cl GEMM via fp32 WMMA: 128 M-tiles x 8 N-tiles, 4 waves/block
    dim3 gridA(NN / 16, (CATOM / 16) / 4);
    cl_wmma_f32_kernel<<<gridA, 128, 0, stream>>>(
        pos, charge, maskp, elem, names, Wf, bf, out_cl);

    // plm pair tensor: one thread per (l,m)
    dim3 gridB(NN / 256, NN);
    plm_pair_kernel<<<gridB, 256, 0, stream>>>(
        pos, uid, W_off, b_off, W_inv, b_inv, W_vm, b_vm, out_plm);
}